// HyperbolicMessagePassing_64622077935753
// MI455X (gfx1250) — compile-verified
//
#include <hip/hip_runtime.h>
#include <hip/hip_bf16.h>
#include <math.h>

typedef __attribute__((ext_vector_type(2))) float v2f;
typedef __attribute__((ext_vector_type(8))) float v8f;

#define D 128
#define LDS_STRIDE 132   // 128 + 4 pad -> rotates banks by 4 per row (64 banks)
#define EPS 1e-8f

// ---------------- kernel 1: zero workspace ----------------
__global__ void hmp_zero_kernel(float* __restrict__ p, int n) {
    int i = blockIdx.x * blockDim.x + threadIdx.x;
    if (i < n) p[i] = 0.0f;
}

// ---------------- kernel 2: edge scatter (feature accumulate + degree count) ----------------
// 1 wave32 per edge; each lane moves float4 (32 lanes * 4 = 128 feats).
__global__ void hmp_scatter_kernel(const float* __restrict__ x,
                                   const int* __restrict__ edge_index,
                                   float* __restrict__ agg,
                                   float* __restrict__ cnt,
                                   int E) {
    int wave = (blockIdx.x * blockDim.x + threadIdx.x) >> 5;  // gfx1250 wave32
    int lane = threadIdx.x & 31;
    if (wave >= E) return;                                    // uniform per wave
    int row = edge_index[wave];       // edge_index[0][e] : destination
    int col = edge_index[E + wave];   // edge_index[1][e] : source
    const float4 v = ((const float4*)(x + (size_t)col * D))[lane];
    float* dst = agg + (size_t)row * D + lane * 4;
    atomicAdd(dst + 0, v.x);
    atomicAdd(dst + 1, v.y);
    atomicAdd(dst + 2, v.z);
    atomicAdd(dst + 3, v.w);
    if (lane == 0) atomicAdd(cnt + row, 1.0f);
}

// ---------------- kernel 3: fused  (agg*scale)@Wm^T + f*bm -> @Wu^T + bu -> row-normalize ----------------
// Block = 256 threads = 8 waves. Block owns 16 nodes; wave w owns output cols [16w,16w+16).
__global__ void __launch_bounds__(256)
hmp_gemm_kernel(const float* __restrict__ agg,
                const float* __restrict__ cnt,
                const float* __restrict__ W_msg, const float* __restrict__ b_msg,
                const float* __restrict__ W_upd, const float* __restrict__ b_upd,
                float* __restrict__ out, int N) {
    __shared__ float scale_s[16];
    __shared__ float f_s[16];
    __shared__ float m_s[16 * LDS_STRIDE];
    __shared__ float o_s[16 * LDS_STRIDE];
    __shared__ float red_s[16 * 16];
    __shared__ float inv_s[16];

    const int tid  = threadIdx.x;
    const int lane = tid & 31;
    const int wave = tid >> 5;          // 0..7
    const int n_base = wave * 16;       // output-column tile
    const int node_base = blockIdx.x * 16;

    const int mloc  = lane & 15;        // A: M index / B: N index
    const int khalf = (lane >> 4) * 2;  // K sub-pair selected by lane half
    const int half  = lane >> 4;

    if (tid < 16) {
        int n = node_base + tid;
        float c = (n < N) ? cnt[n] : 0.0f;
        float s = 1.0f / (c + EPS);
        scale_s[tid] = s;
        f_s[tid]     = c * s;           // count/(count+eps) for bias term
    }
    __syncthreads();

    // ---- GEMM1: m = (agg*scale) @ W_msg^T   (K = 128, step 4) ----
    {
        int arow = node_base + mloc; if (arow >= N) arow = N - 1;
        const float* aptr = agg + (size_t)arow * D;
        const float* bptr = W_msg + (size_t)(n_base + mloc) * D;  // W[n][k]
        const float s = scale_s[mloc];
        v8f acc = {};
        #pragma unroll 4
        for (int k = 0; k < D; k += 4) {
            int kb = k + khalf;
            v2f a, b;
            a.x = aptr[kb] * s;  a.y = aptr[kb + 1] * s;
            b.x = bptr[kb];      b.y = bptr[kb + 1];
            acc = __builtin_amdgcn_wmma_f32_16x16x4_f32(
                    false, a, false, b, (short)0, acc, false, false);
        }
        const int ncol = n_base + mloc;
        const float bm = b_msg[ncol];
        #pragma unroll
        for (int v = 0; v < 8; ++v) {
            int m = v + 8 * half;
            m_s[m * LDS_STRIDE + ncol] = acc[v] + f_s[m] * bm;
        }
    }
    __syncthreads();

    // ---- GEMM2: o = m @ W_upd^T ----
    {
        const float* bptr = W_upd + (size_t)(n_base + mloc) * D;
        const float* aptr = m_s + mloc * LDS_STRIDE;
        v8f acc = {};
        #pragma unroll 4
        for (int k = 0; k < D; k += 4) {
            int kb = k + khalf;
            v2f a, b;
            a.x = aptr[kb];  a.y = aptr[kb + 1];
            b.x = bptr[kb];  b.y = bptr[kb + 1];
            acc = __builtin_amdgcn_wmma_f32_16x16x4_f32(
                    false, a, false, b, (short)0, acc, false, false);
        }
        const int ncol = n_base + mloc;
        const float bu = b_upd[ncol];
        #pragma unroll
        for (int v = 0; v < 8; ++v) {
            int m = v + 8 * half;
            o_s[m * LDS_STRIDE + ncol] = acc[v] + bu;
        }
    }
    __syncthreads();

    // ---- row-wise L2 norm: 16 partial-summers per row ----
    {
        int r = tid >> 4, j = tid & 15;
        float s = 0.0f;
        const float* rp = o_s + r * LDS_STRIDE + j * 8;
        #pragma unroll
        for (int k = 0; k < 8; ++k) { float v = rp[k]; s += v * v; }
        red_s[r * 16 + j] = s;
    }
    __syncthreads();
    if (tid < 16) {
        float s = 0.0f;
        #pragma unroll
        for (int j = 0; j < 16; ++j) s += red_s[tid * 16 + j];
        inv_s[tid] = 1.0f / (sqrtf(s) + EPS);
    }
    __syncthreads();

    // ---- normalized store: 2048 elems / 256 threads = 8 each ----
    #pragma unroll
    for (int i = 0; i < 8; ++i) {
        int idx = tid + i * 256;
        int r = idx >> 7, c = idx & 127;
        int node = node_base + r;
        if (node < N)
            out[(size_t)node * D + c] = o_s[r * LDS_STRIDE + c] * inv_s[r];
    }
}

extern "C" void kernel_launch(void* const* d_in, const int* in_sizes, int n_in,
                              void* d_out, int out_size, void* d_ws, size_t ws_size,
                              hipStream_t stream) {
    const float* x      = (const float*)d_in[0];
    const int*   ei     = (const int*)  d_in[1];
    const float* W_msg  = (const float*)d_in[2];
    const float* b_msg  = (const float*)d_in[3];
    const float* W_upd  = (const float*)d_in[4];
    const float* b_upd  = (const float*)d_in[5];
    float* out = (float*)d_out;

    const int N = in_sizes[0] / D;   // 10000
    const int E = in_sizes[1] / 2;   // 640000

    float* agg = (float*)d_ws;                  // [N][128]
    float* cnt = agg + (size_t)N * D;           // [N]

    int zn = N * D + N;
    hmp_zero_kernel<<<(zn + 255) / 256, 256, 0, stream>>>(agg, zn);

    // 8 waves (edges) per 256-thread block
    hmp_scatter_kernel<<<(E + 7) / 8, 256, 0, stream>>>(x, ei, agg, cnt, E);

    hmp_gemm_kernel<<<(N + 15) / 16, 256, 0, stream>>>(agg, cnt, W_msg, b_msg,
                                                       W_upd, b_upd, out, N);
}